// ChunkLayer_3367254360085
// MI455X (gfx1250) — compile-verified
//
#include <hip/hip_runtime.h>
#include <hip/hip_bf16.h>
#include <stdint.h>

// ---------------------------------------------------------------------------
// ChunkLayer: per-row boolean-mask gather of 4KB token rows + zero padding.
// Pure data movement (~190 MB @ 23.3 TB/s ≈ 8.4 us floor). No FLOPs -> no
// WMMA; the CDNA5-specific paths used are wave32 ballot for the mask scan and
// the async global<->LDS B128 DMA path (ASYNCcnt) for the row copies.
// ---------------------------------------------------------------------------

// True vector types (HIP's float4 is a class; builtins need real vectors).
typedef float v4f __attribute__((ext_vector_type(4)));
typedef int   v4i __attribute__((vector_size(16)));

#if defined(__AMDGCN__) &&                                                    \
    __has_builtin(__builtin_amdgcn_global_load_async_to_lds_b128) &&          \
    __has_builtin(__builtin_amdgcn_global_store_async_from_lds_b128)
#define USE_ASYNC_LDS 1
// Pointer-to-AS1-v4i and pointer-to-AS3-v4i (attribute qualifies the pointee).
typedef __attribute__((address_space(1))) v4i* gv4i_p;
typedef __attribute__((address_space(3))) v4i* lv4i_p;
#else
#define USE_ASYNC_LDS 0
#endif

__device__ __forceinline__ void wait_asynccnt0() {
#if defined(__AMDGCN__)
#if __has_builtin(__builtin_amdgcn_s_wait_asynccnt)
  __builtin_amdgcn_s_wait_asynccnt(0);
#else
  asm volatile("s_wait_asynccnt 0" ::: "memory");
#endif
#endif
}

__device__ __forceinline__ unsigned ballot32(int pred) {
#if defined(__AMDGCN__) && __has_builtin(__builtin_amdgcn_ballot_w32)
  return __builtin_amdgcn_ballot_w32(pred != 0);
#else
  return (unsigned)__ballot(pred != 0);
#endif
}

// ---------------------------------------------------------------------------
// Kernel 1: one wave32 per batch row. Stable stream-compaction scan of the
// mask: indices[b*max_len + j] = source position of the j-th kept token.
// 128 ballot iterations per row; negligible cost vs the 190 MB copy.
// ---------------------------------------------------------------------------
__global__ __launch_bounds__(32) void chunk_scan_kernel(
    const int* __restrict__ mask,   // [B, S] (bool promoted to int32)
    int* __restrict__ indices,      // [B, max_len]
    int* __restrict__ counts,       // [B]
    int S, int max_len) {
  const int b = blockIdx.x;
  const int lane = threadIdx.x;           // wave32: 0..31
  const int* m = mask + (size_t)b * S;
  const unsigned lt_mask = (1u << lane) - 1u;

  int base = 0;
  for (int s0 = 0; s0 < S; s0 += 32) {
    const int s = s0 + lane;
    const int pred = (s < S) ? m[s] : 0;
    const unsigned bal = ballot32(pred);
    if (pred) {
      const int pos = base + (int)__popc(bal & lt_mask);
      if (pos < max_len) indices[(size_t)b * max_len + pos] = s;
    }
    base += (int)__popc(bal);
  }
  if (lane == 0) counts[b] = (base < max_len) ? base : max_len;
}

// ---------------------------------------------------------------------------
// Kernel 2: one 256-thread block per output row (b, j). Valid rows: copy the
// 4KB source row via async global->LDS->global B128 DMA (each lane owns one
// 16B slice; one s_wait_asynccnt 0 per wave between load and store). Padding
// rows: store zeros. Every output element is written each call.
// ---------------------------------------------------------------------------
__global__ __launch_bounds__(256) void chunk_gather_kernel(
    const float* __restrict__ hs,     // [B, S, D]
    const int* __restrict__ indices,  // [B, max_len]
    const int* __restrict__ counts,   // [B]
    float* __restrict__ out,          // [B, max_len, D]
    int S, int D, int max_len) {
  __shared__ v4f buf[256];            // 4 KB staging (one 16B slice per lane)

  const int row = blockIdx.x;         // flat (b * max_len + j)
  const int b = row / max_len;
  const int j = row - b * max_len;
  const int t = threadIdx.x;
  const int n4 = D >> 2;              // 16B chunks per row (256 for D=1024)

  v4f* __restrict__ orow = reinterpret_cast<v4f*>(out + (size_t)row * D);

  if (j < counts[b]) {
    const int src = indices[row];
    const v4f* __restrict__ irow = reinterpret_cast<const v4f*>(
        hs + ((size_t)b * S + (size_t)src) * D);
    for (int i = t; i < n4; i += (int)blockDim.x) {
#if USE_ASYNC_LDS
      gv4i_p g_src = (gv4i_p)(irow + i);   // drop const + addrspacecast to AS1
      gv4i_p g_dst = (gv4i_p)(orow + i);
      lv4i_p l_buf = (lv4i_p)(&buf[t]);    // AS3 slice owned by this lane
      // memory -> LDS (ASYNCcnt++)
      __builtin_amdgcn_global_load_async_to_lds_b128(g_src, l_buf, 0, 0);
      wait_asynccnt0();               // load landed in this wave's LDS slice
      // LDS -> memory (ASYNCcnt++)
      __builtin_amdgcn_global_store_async_from_lds_b128(g_dst, l_buf, 0, 0);
      wait_asynccnt0();               // WAR guard before LDS slot reuse
#else
      const v4f v = __builtin_nontemporal_load(irow + i);
      __builtin_nontemporal_store(v, orow + i);
#endif
    }
  } else {
    const v4f z = {0.f, 0.f, 0.f, 0.f};
    for (int i = t; i < n4; i += (int)blockDim.x) {
      __builtin_nontemporal_store(z, orow + i);
    }
  }
}

// ---------------------------------------------------------------------------
// Host launcher
// ---------------------------------------------------------------------------
extern "C" void kernel_launch(void* const* d_in, const int* in_sizes, int n_in,
                              void* d_out, int out_size, void* d_ws, size_t ws_size,
                              hipStream_t stream) {
  const float* hs  = (const float*)d_in[0];  // [B, S, D] float32
  const int* mask  = (const int*)d_in[1];    // [B, S] bool -> int32 per harness
  float* out       = (float*)d_out;          // [B, max_len, D] float32

  const int B = 8;                                   // fixed by reference
  const int S = in_sizes[1] / B;                     // 4096
  const int D = in_sizes[0] / in_sizes[1];           // 1024
  const int max_len = out_size / (B * D);            // data-dependent pad len

  int* indices = (int*)d_ws;                         // [B, max_len]
  int* counts  = indices + (size_t)B * max_len;      // [B]

  chunk_scan_kernel<<<B, 32, 0, stream>>>(mask, indices, counts, S, max_len);

  const int rows = B * max_len;
  chunk_gather_kernel<<<rows, 256, 0, stream>>>(hs, indices, counts, out, S, D,
                                                max_len);
}